// GAT_635655160272
// MI455X (gfx1250) — compile-verified
//
#include <hip/hip_runtime.h>

#define N_NODES 50000
#define N_EDGES 800000
#define HEADS   4
#define FDIM    64
#define DOUT    256   // HEADS * FDIM

typedef __attribute__((ext_vector_type(2))) float v2f;
typedef __attribute__((ext_vector_type(8))) float v8f;

// ---------- order-preserving float <-> uint for atomic max ----------
__device__ __forceinline__ unsigned fflip(float f) {
    unsigned u = __float_as_uint(f);
    return (u & 0x80000000u) ? ~u : (u | 0x80000000u);
}
__device__ __forceinline__ float funflip(unsigned u) {
    unsigned v = (u & 0x80000000u) ? (u ^ 0x80000000u) : ~u;
    return __uint_as_float(v);
}

// ---------- WMMA fp32 GEMM: out[M x 256] = X[M x K] * W[K x 256] ----------
// grid.x = M/16 row tiles; block = 128 threads = 4 waves; wave w -> cols [w*64, w*64+64)
__global__ __launch_bounds__(128) void gemm_wmma_f32(
    const float* __restrict__ X, const float* __restrict__ W,
    float* __restrict__ out, int K)
{
    const int lane  = threadIdx.x & 31;
    const int wave  = threadIdx.x >> 5;
    const int mBase = blockIdx.x * 16;
    const int nBase = wave * 64;
    const int hb    = lane >> 4;   // 0: lanes 0-15, 1: lanes 16-31
    const int l15   = lane & 15;

    const int    aRow = mBase + l15;
    const float* aPtr = X + (size_t)aRow * K + hb * 2;

    v8f acc0 = {}, acc1 = {}, acc2 = {}, acc3 = {};

    for (int k = 0; k < K; k += 4) {
        // A fragment: lane<16 holds {A[m][k],A[m][k+1]}, lane>=16 {A[m][k+2],A[m][k+3]}
        v2f a;
        a.x = aPtr[k];
        a.y = aPtr[k + 1];
        // B fragments (rows k..k+3 of W), mirrored half-lane layout
        const float* bRow0 = W + (size_t)(k + hb * 2) * DOUT + nBase + l15;
        const float* bRow1 = bRow0 + DOUT;
        v2f b0, b1, b2, b3;
        b0.x = bRow0[0];  b0.y = bRow1[0];
        b1.x = bRow0[16]; b1.y = bRow1[16];
        b2.x = bRow0[32]; b2.y = bRow1[32];
        b3.x = bRow0[48]; b3.y = bRow1[48];
        acc0 = __builtin_amdgcn_wmma_f32_16x16x4_f32(false, a, false, b0, (short)0, acc0, false, false);
        acc1 = __builtin_amdgcn_wmma_f32_16x16x4_f32(false, a, false, b1, (short)0, acc1, false, false);
        acc2 = __builtin_amdgcn_wmma_f32_16x16x4_f32(false, a, false, b2, (short)0, acc2, false, false);
        acc3 = __builtin_amdgcn_wmma_f32_16x16x4_f32(false, a, false, b3, (short)0, acc3, false, false);
    }

    // C/D layout: element i -> row = mBase + i + hb*8, col = nBase + tile*16 + l15
    float* o = out + (size_t)(mBase + hb * 8) * DOUT + nBase + l15;
    #pragma unroll
    for (int i = 0; i < 8; ++i) {
        o[(size_t)i * DOUT + 0]  = acc0[i];
        o[(size_t)i * DOUT + 16] = acc1[i];
        o[(size_t)i * DOUT + 32] = acc2[i];
        o[(size_t)i * DOUT + 48] = acc3[i];
    }
}

// ---------- el/er: per (node, head) dot of feat slice with attn vectors ----------
__global__ void attn_dots(const float* __restrict__ feat,
                          const float* __restrict__ al,
                          const float* __restrict__ ar,
                          float* __restrict__ el, float* __restrict__ er)
{
    int idx = blockIdx.x * blockDim.x + threadIdx.x;   // n*HEADS + h
    if (idx >= N_NODES * HEADS) return;
    int h = idx & (HEADS - 1);
    const float4* f = (const float4*)(feat + (size_t)idx * FDIM); // idx*64 == n*256 + h*64
    const float4* a = (const float4*)(al + h * FDIM);
    const float4* b = (const float4*)(ar + h * FDIM);
    float sl = 0.f, sr = 0.f;
    #pragma unroll
    for (int i = 0; i < FDIM / 4; ++i) {
        float4 fv = f[i], av = a[i], bv = b[i];
        sl += fv.x * av.x + fv.y * av.y + fv.z * av.z + fv.w * av.w;
        sr += fv.x * bv.x + fv.y * bv.y + fv.z * bv.z + fv.w * bv.w;
    }
    el[idx] = sl;
    er[idx] = sr;
}

// ---------- edge pass 1: leaky_relu(el[src]+er[dst]) and segment max ----------
__global__ void edge_max(const int* __restrict__ src, const int* __restrict__ dst,
                         const float* __restrict__ el, const float* __restrict__ er,
                         float* __restrict__ ebuf, unsigned* __restrict__ mflip)
{
    int e = blockIdx.x * blockDim.x + threadIdx.x;
    if (e >= N_EDGES) return;
    int s = src[e], d = dst[e];
    float4 a = *(const float4*)(el + (size_t)s * HEADS);
    float4 b = *(const float4*)(er + (size_t)d * HEADS);
    float v0 = a.x + b.x, v1 = a.y + b.y, v2 = a.z + b.z, v3 = a.w + b.w;
    v0 = v0 > 0.f ? v0 : 0.2f * v0;
    v1 = v1 > 0.f ? v1 : 0.2f * v1;
    v2 = v2 > 0.f ? v2 : 0.2f * v2;
    v3 = v3 > 0.f ? v3 : 0.2f * v3;
    *(float4*)(ebuf + (size_t)e * HEADS) = make_float4(v0, v1, v2, v3);
    unsigned* mp = mflip + (size_t)d * HEADS;
    atomicMax(&mp[0], fflip(v0));
    atomicMax(&mp[1], fflip(v1));
    atomicMax(&mp[2], fflip(v2));
    atomicMax(&mp[3], fflip(v3));
}

// ---------- edge pass 2: a = exp(e - m[dst]); segment sum ----------
__global__ void edge_exp_sum(const int* __restrict__ dst,
                             float* __restrict__ ebuf,
                             const unsigned* __restrict__ mflip,
                             float* __restrict__ ssum)
{
    int e = blockIdx.x * blockDim.x + threadIdx.x;
    if (e >= N_EDGES) return;
    int d = dst[e];
    float4 ev = *(const float4*)(ebuf + (size_t)e * HEADS);
    uint4  mu = *(const uint4*)(mflip + (size_t)d * HEADS);
    float a0 = expf(ev.x - funflip(mu.x));
    float a1 = expf(ev.y - funflip(mu.y));
    float a2 = expf(ev.z - funflip(mu.z));
    float a3 = expf(ev.w - funflip(mu.w));
    *(float4*)(ebuf + (size_t)e * HEADS) = make_float4(a0, a1, a2, a3);
    float* sp = ssum + (size_t)d * HEADS;
    atomicAdd(&sp[0], a0);
    atomicAdd(&sp[1], a1);
    atomicAdd(&sp[2], a2);
    atomicAdd(&sp[3], a3);
}

// ---------- edge pass 3: acc[dst] += feat[src] * (a / s[dst]) ----------
// one block per edge; thread t -> head t>>6, feature t&63
__global__ __launch_bounds__(256) void edge_aggregate(
    const int* __restrict__ src, const int* __restrict__ dst,
    const float* __restrict__ feat, const float* __restrict__ ebuf,
    const float* __restrict__ ssum, float* __restrict__ acc)
{
    int e = blockIdx.x;
    int t = threadIdx.x;
    int h = t >> 6;
    int s = src[e], d = dst[e];
    float alpha = ebuf[(size_t)e * HEADS + h] / ssum[(size_t)d * HEADS + h];
    float msg   = feat[(size_t)s * DOUT + t] * alpha;
    atomicAdd(&acc[(size_t)d * DOUT + t], msg);
}

// ---------- ELU ----------
__global__ void elu_k(const float* __restrict__ in, float* __restrict__ out, int n)
{
    int i = blockIdx.x * blockDim.x + threadIdx.x;
    if (i >= n) return;
    float v = in[i];
    out[i] = v > 0.f ? v : expm1f(v);
}

// ---------- mean over heads: [N,4,64] -> [N,64] ----------
__global__ void head_mean(const float* __restrict__ acc, float* __restrict__ out)
{
    int i = blockIdx.x * blockDim.x + threadIdx.x;   // n*64 + f
    if (i >= N_NODES * FDIM) return;
    int n = i >> 6, f = i & 63;
    const float* p = acc + (size_t)n * DOUT + f;
    out[i] = 0.25f * (p[0] + p[64] + p[128] + p[192]);
}

extern "C" void kernel_launch(void* const* d_in, const int* in_sizes, int n_in,
                              void* d_out, int out_size, void* d_ws, size_t ws_size,
                              hipStream_t stream)
{
    (void)in_sizes; (void)n_in; (void)out_size; (void)ws_size;

    const float* h   = (const float*)d_in[0];
    const float* W0  = (const float*)d_in[1];
    const float* al0 = (const float*)d_in[2];
    const float* ar0 = (const float*)d_in[3];
    const float* W1  = (const float*)d_in[4];
    const float* al1 = (const float*)d_in[5];
    const float* ar1 = (const float*)d_in[6];
    const float* W2  = (const float*)d_in[7];
    const float* al2 = (const float*)d_in[8];
    const float* ar2 = (const float*)d_in[9];
    const int*   src = (const int*)d_in[10];
    const int*   dst = (const int*)d_in[11];
    float* logits    = (float*)d_out;

    // workspace carve-up (all offsets 256B aligned)
    char* ws = (char*)d_ws;
    const size_t nodeMat = (size_t)N_NODES * DOUT * sizeof(float);   // 51.2 MB
    const size_t nodeVec = (size_t)N_NODES * HEADS * sizeof(float);  // 0.8 MB
    const size_t edgeMat = (size_t)N_EDGES * HEADS * sizeof(float);  // 12.8 MB
    float*    feat = (float*)ws;             ws += nodeMat;
    float*    accb = (float*)ws;             ws += nodeMat;
    float*    xbuf = (float*)ws;             ws += nodeMat;
    float*    el   = (float*)ws;             ws += nodeVec;
    float*    er   = (float*)ws;             ws += nodeVec;
    unsigned* mfl  = (unsigned*)ws;          ws += nodeVec;
    float*    ssum = (float*)ws;             ws += nodeVec;
    float*    ebuf = (float*)ws;             ws += edgeMat;

    const dim3 gemmGrid(N_NODES / 16), gemmBlk(128);
    const int nh      = N_NODES * HEADS;
    const int nhBlks  = (nh + 255) / 256;
    const int eBlks   = (N_EDGES + 255) / 256;
    const int nmElems = N_NODES * DOUT;
    const int nmBlks  = (nmElems + 255) / 256;

    auto run_layer = [&](const float* xin, int K, const float* Wm,
                         const float* al_, const float* ar_) {
        gemm_wmma_f32<<<gemmGrid, gemmBlk, 0, stream>>>(xin, Wm, feat, K);
        attn_dots<<<nhBlks, 256, 0, stream>>>(feat, al_, ar_, el, er);
        hipMemsetAsync(mfl,  0, nodeVec, stream);   // fflip^-1(0) == smallest
        hipMemsetAsync(ssum, 0, nodeVec, stream);
        hipMemsetAsync(accb, 0, nodeMat, stream);
        edge_max<<<eBlks, 256, 0, stream>>>(src, dst, el, er, ebuf, mfl);
        edge_exp_sum<<<eBlks, 256, 0, stream>>>(dst, ebuf, mfl, ssum);
        edge_aggregate<<<N_EDGES, 256, 0, stream>>>(src, dst, feat, ebuf, ssum, accb);
    };

    // layer 0: Din = 128
    run_layer(h, 128, W0, al0, ar0);
    elu_k<<<nmBlks, 256, 0, stream>>>(accb, xbuf, nmElems);
    // layer 1: Din = 256
    run_layer(xbuf, 256, W1, al1, ar1);
    elu_k<<<nmBlks, 256, 0, stream>>>(accb, xbuf, nmElems);
    // layer 2: Din = 256, then mean over heads
    run_layer(xbuf, 256, W2, al2, ar2);
    head_mean<<<(N_NODES * FDIM + 255) / 256, 256, 0, stream>>>(accb, logits);
}